// GATv2_18433999635058
// MI455X (gfx1250) — compile-verified
//
#include <hip/hip_runtime.h>
#include <math.h>

typedef __attribute__((ext_vector_type(2))) float v2f;
typedef __attribute__((ext_vector_type(8))) float v8f;

#define HEADS 8

#if defined(__gfx1250__) && __has_builtin(__builtin_amdgcn_wmma_f32_16x16x4_f32)
#define USE_WMMA_F32 1
#else
#define USE_WMMA_F32 0
#endif

// ---- orderable-uint encoding for float atomic max ----
__device__ __forceinline__ unsigned enc_ord(float f) {
  unsigned b = __float_as_uint(f);
  return (b & 0x80000000u) ? ~b : (b | 0x80000000u);
}
__device__ __forceinline__ float dec_ord(unsigned u) {
  unsigned b = (u & 0x80000000u) ? (u ^ 0x80000000u) : ~u;
  return __uint_as_float(b);
}
// enc(-inf) == 0x007FFFFF

// ------------------------------------------------------------------
// Generic f32 GEMM via V_WMMA_F32_16X16X4_F32. One 16x16 tile / wave.
// A: [M,Kin] row-major, W: [Kin,Kout] row-major, C: [M,Kout].
// M % 16 == 0, Kin % 4 == 0, Kout % 16 == 0 (all hold for this problem).
// ------------------------------------------------------------------
__global__ void gemm_wmma(const float* __restrict__ A, const float* __restrict__ W,
                          float* __restrict__ C, int M, int Kin, int Kout) {
  int wid = blockIdx.x * (blockDim.x >> 5) + (threadIdx.x >> 5);
  int tilesN = Kout >> 4;
  int tiles  = (M >> 4) * tilesN;
  if (wid >= tiles) return;                 // wave-uniform: EXEC stays all-ones
  int tm = wid / tilesN, tn = wid % tilesN;
  int lane = threadIdx.x & 31;
  int grp  = lane >> 4;                     // lane half: selects K pair / M half
  int ln   = lane & 15;

#if USE_WMMA_F32
  const float* Arow = A + (size_t)(tm * 16 + ln) * Kin;   // A row for this lane
  const float* Wcol = W + (size_t)tn * 16 + ln;           // B column for this lane
  v8f acc = {};
  for (int k = 0; k < Kin; k += 4) {
    v2f a, b;
    a[0] = Arow[k + 2 * grp];
    a[1] = Arow[k + 2 * grp + 1];
    b[0] = Wcol[(size_t)(k + 2 * grp) * Kout];
    b[1] = Wcol[(size_t)(k + 2 * grp + 1) * Kout];
    acc = __builtin_amdgcn_wmma_f32_16x16x4_f32(false, a, false, b,
                                                (short)0, acc, false, false);
  }
  float* Cout = C + (size_t)(tm * 16 + grp * 8) * Kout + (size_t)tn * 16 + ln;
#pragma unroll
  for (int v = 0; v < 8; ++v) Cout[(size_t)v * Kout] = acc[v];
#else
  float acc[8] = {};
  for (int k = 0; k < Kin; ++k) {
    float bv = W[(size_t)k * Kout + tn * 16 + ln];
#pragma unroll
    for (int v = 0; v < 8; ++v)
      acc[v] += A[(size_t)(tm * 16 + grp * 8 + v) * Kin + k] * bv;
  }
  float* Cout = C + (size_t)(tm * 16 + grp * 8) * Kout + (size_t)tn * 16 + ln;
#pragma unroll
  for (int v = 0; v < 8; ++v) Cout[(size_t)v * Kout] = acc[v];
#endif
}

// ------------------------------------------------------------------
__global__ void fill_u32(unsigned* __restrict__ p, unsigned v, long long n) {
  long long i = blockIdx.x * (long long)blockDim.x + threadIdx.x;
  long long stride = (long long)gridDim.x * blockDim.x;
  for (; i < n; i += stride) p[i] = v;
}

__device__ __forceinline__ void edge_sd(const int* __restrict__ srcI,
                                        const int* __restrict__ dstI,
                                        int e, int E, int& s, int& d) {
  if (e < E) { s = srcI[e]; d = dstI[e]; }
  else       { s = d = e - E; }            // self-loops appended
}

// Pass 1: logits[e,h] = sum_c leaky_relu(xl[s,h,c]+xr[d,h,c], 0.2) * att[h,c]
//         atomic max into lmax[d,h]
__global__ void edge_logits_kernel(const float* __restrict__ xl, const float* __restrict__ xr,
                                   const float* __restrict__ att,
                                   const int* __restrict__ srcI, const int* __restrict__ dstI,
                                   int E, int Etot, int C,
                                   float* __restrict__ logits, unsigned* __restrict__ lmax) {
  long long idx = blockIdx.x * (long long)blockDim.x + threadIdx.x;
  if (idx >= (long long)Etot * HEADS) return;
  int e = (int)(idx >> 3), h = (int)(idx & 7);
  int s, d; edge_sd(srcI, dstI, e, E, s, d);
  const float4* pl = (const float4*)(xl + ((size_t)s * HEADS + h) * C);
  const float4* pr = (const float4*)(xr + ((size_t)d * HEADS + h) * C);
  const float4* pa = (const float4*)(att + (size_t)h * C);
  float acc = 0.f;
  int n4 = C >> 2;
  for (int i = 0; i < n4; ++i) {
    float4 a = pl[i], b = pr[i], w = pa[i];
    float v;
    v = a.x + b.x; v = v > 0.f ? v : 0.2f * v; acc += v * w.x;
    v = a.y + b.y; v = v > 0.f ? v : 0.2f * v; acc += v * w.y;
    v = a.z + b.z; v = v > 0.f ? v : 0.2f * v; acc += v * w.z;
    v = a.w + b.w; v = v > 0.f ? v : 0.2f * v; acc += v * w.w;
  }
  logits[idx] = acc;
  atomicMax(lmax + (size_t)d * HEADS + h, enc_ord(acc));
}

// Pass 2 (fused exp + aggregate, unnormalized):
//   ex = exp(logit - max);  denom[d,h] += ex;  out[d,h,:] += ex * xl[s,h,:]
// Normalization by denom happens in the node-level epilogues.
__global__ void edge_accum_kernel(const float* __restrict__ xl,
                                  const int* __restrict__ srcI, const int* __restrict__ dstI,
                                  int E, int Etot, int C,
                                  const float* __restrict__ logits,
                                  const unsigned* __restrict__ lmax,
                                  float* __restrict__ denom,
                                  float* __restrict__ out) {
  long long idx = blockIdx.x * (long long)blockDim.x + threadIdx.x;
  if (idx >= (long long)Etot * HEADS) return;
  int e = (int)(idx >> 3), h = (int)(idx & 7);
  int s, d; edge_sd(srcI, dstI, e, E, s, d);
  float m = dec_ord(lmax[(size_t)d * HEADS + h]);
  float ex = expf(logits[idx] - m);
  atomicAdd(denom + (size_t)d * HEADS + h, ex);
  const float4* pm = (const float4*)(xl + ((size_t)s * HEADS + h) * C);
  float* po = out + ((size_t)d * HEADS + h) * C;
  int n4 = C >> 2;
  for (int i = 0; i < n4; ++i) {
    float4 m4 = pm[i];
    atomicAdd(po + 4 * i + 0, ex * m4.x);
    atomicAdd(po + 4 * i + 1, ex * m4.y);
    atomicAdd(po + 4 * i + 2, ex * m4.z);
    atomicAdd(po + 4 * i + 3, ex * m4.w);
  }
}

// Layer-1 epilogue: h = elu(accum/denom + b1), elementwise over N*256 (C=32)
__global__ void norm_elu_bias_kernel(float* __restrict__ h, const float* __restrict__ denom,
                                     const float* __restrict__ b, long long total) {
  long long i = blockIdx.x * (long long)blockDim.x + threadIdx.x;
  long long stride = (long long)gridDim.x * blockDim.x;
  for (; i < total; i += stride) {
    int rem = (int)(i & 255);                 // i % 256
    long long n = i >> 8;                     // i / 256
    int hd = rem >> 5;                        // rem / 32
    float v = h[i] / denom[n * HEADS + hd] + b[rem];
    h[i] = v > 0.f ? v : (expf(v) - 1.f);
  }
}

// Layer-2 epilogue: v = accum/denom + b2, then in-place log_softmax over 128 (C=16)
__global__ void finalize_logsoftmax(float* __restrict__ out, const float* __restrict__ denom,
                                    const float* __restrict__ b2) {
  int n = blockIdx.x;
  int t = threadIdx.x;
  int hd = t >> 4;                            // head = t / 16
  float v = out[(size_t)n * 128 + t] / denom[(size_t)n * HEADS + hd] + b2[t];
  __shared__ float sm[128];
  sm[t] = v; __syncthreads();
  for (int sft = 64; sft > 0; sft >>= 1) { if (t < sft) sm[t] = fmaxf(sm[t], sm[t + sft]); __syncthreads(); }
  float mx = sm[0]; __syncthreads();
  sm[t] = expf(v - mx); __syncthreads();
  for (int sft = 64; sft > 0; sft >>= 1) { if (t < sft) sm[t] += sm[t + sft]; __syncthreads(); }
  float lse = logf(sm[0]);
  out[(size_t)n * 128 + t] = v - mx - lse;
}

// ------------------------------------------------------------------
extern "C" void kernel_launch(void* const* d_in, const int* in_sizes, int n_in,
                              void* d_out, int out_size, void* d_ws, size_t ws_size,
                              hipStream_t stream) {
  (void)n_in; (void)out_size; (void)ws_size;
  const float* x    = (const float*)d_in[0];
  const int*   ei   = (const int*)d_in[1];   // int32 (JAX x64 disabled)
  const float* W1l  = (const float*)d_in[2];
  const float* W1r  = (const float*)d_in[3];
  const float* att1 = (const float*)d_in[4];
  const float* b1   = (const float*)d_in[5];
  const float* W2l  = (const float*)d_in[6];
  const float* W2r  = (const float*)d_in[7];
  const float* att2 = (const float*)d_in[8];
  const float* b2   = (const float*)d_in[9];

  int N = in_sizes[0] / 128;
  int E = in_sizes[1] / 2;
  int Etot = E + N;
  const int* srcI = ei;
  const int* dstI = ei + E;

  // workspace layout (floats)
  float* ws    = (float*)d_ws;
  float* xl    = ws;                              // N*256
  float* xr    = xl + (size_t)N * 256;            // N*256
  float* hbuf  = xr + (size_t)N * 256;            // N*256 (layer1 accum -> elu -> h)
  float* lg    = hbuf + (size_t)N * 256;          // Etot*8
  unsigned* lmax = (unsigned*)(lg + (size_t)Etot * 8);  // N*8
  float* denom = (float*)(lmax + (size_t)N * 8);        // N*8
  float* outAcc = (float*)d_out;                  // N*128, layer2 accum + final

  dim3 blk(256);
  long long eh = (long long)Etot * HEADS;
  int egrid = (int)((eh + 255) / 256);
  const unsigned ENC_NEG_INF = 0x007FFFFFu;       // enc_ord(-inf)

  // ---------------- Layer 1: in=128 -> 8x32 ----------------
  int tiles1 = (N / 16) * (256 / 16);
  gemm_wmma<<<(tiles1 + 7) / 8, blk, 0, stream>>>(x, W1l, xl, N, 128, 256);
  gemm_wmma<<<(tiles1 + 7) / 8, blk, 0, stream>>>(x, W1r, xr, N, 128, 256);
  fill_u32<<<1024, blk, 0, stream>>>(lmax, ENC_NEG_INF, (long long)N * 8);
  fill_u32<<<1024, blk, 0, stream>>>((unsigned*)denom, 0u, (long long)N * 8);
  fill_u32<<<2048, blk, 0, stream>>>((unsigned*)hbuf, 0u, (long long)N * 256);
  edge_logits_kernel<<<egrid, blk, 0, stream>>>(xl, xr, att1, srcI, dstI, E, Etot, 32, lg, lmax);
  edge_accum_kernel<<<egrid, blk, 0, stream>>>(xl, srcI, dstI, E, Etot, 32, lg, lmax, denom, hbuf);
  norm_elu_bias_kernel<<<2048, blk, 0, stream>>>(hbuf, denom, b1, (long long)N * 256);

  // ---------------- Layer 2: in=256 -> 8x16 ----------------
  int tiles2 = (N / 16) * (128 / 16);
  gemm_wmma<<<(tiles2 + 7) / 8, blk, 0, stream>>>(hbuf, W2l, xl, N, 256, 128);
  gemm_wmma<<<(tiles2 + 7) / 8, blk, 0, stream>>>(hbuf, W2r, xr, N, 256, 128);
  fill_u32<<<1024, blk, 0, stream>>>(lmax, ENC_NEG_INF, (long long)N * 8);
  fill_u32<<<1024, blk, 0, stream>>>((unsigned*)denom, 0u, (long long)N * 8);
  fill_u32<<<2048, blk, 0, stream>>>((unsigned*)outAcc, 0u, (long long)N * 128);
  edge_logits_kernel<<<egrid, blk, 0, stream>>>(xl, xr, att2, srcI, dstI, E, Etot, 16, lg, lmax);
  edge_accum_kernel<<<egrid, blk, 0, stream>>>(xl, srcI, dstI, E, Etot, 16, lg, lmax, denom, outAcc);
  finalize_logsoftmax<<<N, dim3(128), 0, stream>>>(outAcc, denom, b2);
}